// MyMaxPool2D_36189394436997
// MI455X (gfx1250) — compile-verified
//
#include <hip/hip_runtime.h>
#include <hip/hip_bf16.h>

// Max pool 2x2, stride 2, VALID, NCHW fp32.
// Input : (16, 64, 512, 512)  -> planes P = 16*64 = 1024, H=W=512
// Output: (16, 64, 256, 256)  -> 67,108,864 floats
//
// Bandwidth-bound: ~1.34 GB traffic, floor ~58 us @ 23.3 TB/s.
// Strategy: 2 outputs per thread.
//   - lane loads float4 of input row 2i   (cols 2j..2j+3)   [b128, NT]
//   - lane loads float4 of input row 2i+1 (cols 2j..2j+3)   [b128, NT]
//   - 2 outputs = pairwise max reduction (v_max3_f32 folding)
//   - store float2                                           [b64,  NT]
// Consecutive lanes are byte-contiguous for every memory instruction
// (dense 512B/wave loads, 256B/wave stores, wave32).

typedef float v4f __attribute__((ext_vector_type(4)));
typedef float v2f __attribute__((ext_vector_type(2)));

__global__ void __launch_bounds__(256)
maxpool2x2_nt_kernel(const float* __restrict__ in,
                     float* __restrict__ out,
                     long long n_threads)
{
    long long t = (long long)blockIdx.x * blockDim.x + threadIdx.x;
    if (t >= n_threads) return;

    // 128 threads per output row of 256 elements (2 outputs each).
    // row = global output row index in [0, P*256); all extents are pow2.
    const int  jj  = ((int)(t & 127)) << 1;   // output column (even)
    long long  row = t >> 7;                  // (plane*256 + i)
    const long long plane = row >> 8;         // plane in [0, 1024)
    const int  i   = (int)(row & 255);        // output row within plane

    // Input base: plane*512*512 + (2*i)*512 + 2*jj   (floats)
    const long long ibase = (plane << 18) + ((long long)i << 10) + ((long long)jj << 1);

    const v4f a = __builtin_nontemporal_load((const v4f*)(in + ibase));        // row 2i
    const v4f b = __builtin_nontemporal_load((const v4f*)(in + ibase + 512));  // row 2i+1

    v2f o;
    o.x = fmaxf(fmaxf(a.x, a.y), fmaxf(b.x, b.y));
    o.y = fmaxf(fmaxf(a.z, a.w), fmaxf(b.z, b.w));

    // Output base: row*256 + jj (floats), 8B aligned.
    __builtin_nontemporal_store(o, (v2f*)(out + (row << 8) + jj));
}

extern "C" void kernel_launch(void* const* d_in, const int* in_sizes, int n_in,
                              void* d_out, int out_size, void* d_ws, size_t ws_size,
                              hipStream_t stream)
{
    (void)in_sizes; (void)n_in; (void)d_ws; (void)ws_size;

    const float* x = (const float*)d_in[0];
    float*       y = (float*)d_out;

    // out_size = 16*64*256*256 = 67,108,864 ; 2 outputs per thread.
    const long long n_threads = (long long)out_size / 2;   // 33,554,432
    const int block = 256;                                  // 8 waves (wave32)
    const long long grid = (n_threads + block - 1) / block; // 131,072 blocks

    maxpool2x2_nt_kernel<<<(unsigned int)grid, block, 0, stream>>>(x, y, n_threads);
}